// SpectralTokenizer_62328565399739
// MI455X (gfx1250) — compile-verified
//
#include <hip/hip_runtime.h>
#include <hip/hip_bf16.h>
#include <math.h>

// ---------------- problem constants ----------------
#define T_LEN     38400
#define N_FFT     256
#define HOP       64
#define PAD       128           // N_FFT/2
#define N_FRAMES  601           // 1 + T/HOP
#define N_BLK     19            // ceil(601/32) frame blocks of 32
#define BC        1024          // B*C = 16*64
#define NBINS     101           // bins 2..102 inclusive
#define NCOLS     202           // cos+sin interleaved per bin
#define NCOLS_PAD 224           // 14 tiles of 16 -> 7 N-tile pairs
#define N_NPAIRS  7
#define KSTRIDE   258           // row stride in LDS (bank spread)
#define D_TOKEN   128
#define INV_SCALE (1.0f / (16.0f * 601.0f))   // /sqrt(256) and /n_frames

typedef float v2f __attribute__((ext_vector_type(2)));
typedef float v8f __attribute__((ext_vector_type(8)));

// Issue async gathers (global -> LDS) for one 32-frame block.
// Each wave's loads are tracked by its own ASYNCcnt.
static __device__ __forceinline__ void stage_frames_async(
    float* __restrict__ buf, const float* __restrict__ xs, int blk, int tid)
{
    const int f0 = blk * 32;
    for (int idx = tid; idx < 32 * N_FFT; idx += 256) {
        const int fr = idx >> 8;
        const int t  = idx & 255;
        const int f  = f0 + fr;
        float* dst = &buf[fr * KSTRIDE + t];
        if (f < N_FRAMES) {
            int s = f * HOP + t - PAD;
            if (s < 0) s = -s;
            else if (s >= T_LEN) s = 2 * T_LEN - 2 - s;
            const float* src = &xs[s];
            // generic LDS pointer: low 32 bits == DS-relative byte address
            unsigned lds_off = (unsigned)(uintptr_t)dst;
            asm volatile("global_load_async_to_lds_b32 %0, %1, off"
                         :: "v"(lds_off), "v"(src) : "memory");
        } else {
            *dst = 0.0f;   // zero-fill frames 601..607 (last block)
        }
    }
}

// ============================================================
// Stage 1: per-signal STFT magnitude mean via fp32 WMMA DFT.
//   grid = 1024 blocks (one per (b,c) signal), 256 threads (8 waves)
//   2x2 register blocking: 32 frames x 32 basis cols per wave work-item
//   -> 4 LDS fragment loads per 4 wmmas (256 B/wmma, matches LDS BW)
//   double-buffered frame staging via GLOBAL_LOAD_ASYNC_TO_LDS_B32
//   dynamic LDS: basisT[224][258] | frames0/1[32][258] | win[256] | acc[104]
// ============================================================
__global__ __launch_bounds__(256) void stft_magmean_wmma_kernel(
    const float* __restrict__ x, float* __restrict__ mag_mean)
{
    extern __shared__ float smem[];
    float* basisT  = smem;                          // 224*258 (col-major [col][t])
    float* frames0 = basisT + NCOLS_PAD * KSTRIDE;  // 32*258
    float* frames1 = frames0 + 32 * KSTRIDE;        // 32*258
    float* win     = frames1 + 32 * KSTRIDE;        // 256
    float* acc     = win + 256;                     // 104 (101 used)

    const int tid  = threadIdx.x;
    const int bc   = blockIdx.x;
    const int lane = tid & 31;
    const int wave = tid >> 5;

    // ---------- one-time init ----------
    win[tid] = 0.5f * (1.0f - cosf(6.2831853071795864769f * (float)tid / 256.0f));
    if (tid < 104) acc[tid] = 0.0f;
    __syncthreads();

    // basis col 2j = win[t]*cos(2*pi*(j+2)*t/256), col 2j+1 = win[t]*sin(...)
    // (window folded into the basis; frames are then raw samples)
    if (tid < NCOLS) {
        const int  j     = tid >> 1;
        const bool iscos = (tid & 1) == 0;
        const double th  = -6.283185307179586476925286766559 * (double)(j + 2) / 256.0;
        const double ct  = cos(th), st = sin(th);
        double cr = 1.0, ci = 0.0;
        float* row = basisT + tid * KSTRIDE;
        for (int t = 0; t < N_FFT; ++t) {
            row[t] = (iscos ? (float)cr : (float)ci) * win[t];
            double nr = cr * ct - ci * st;
            ci = cr * st + ci * ct;
            cr = nr;
        }
        row[256] = 0.0f; row[257] = 0.0f;
    } else if (tid < NCOLS_PAD) {
        float* row = basisT + tid * KSTRIDE;        // zero-pad cols 202..223
        for (int t = 0; t < KSTRIDE; ++t) row[t] = 0.0f;
    }

    const float* xs = x + (size_t)bc * T_LEN;

    // WMMA fragment addressing (wave32, V_WMMA_F32_16X16X4_F32):
    //  A 16x4:  lane m = lane&15, k-pair offset = 2*(lane>>4)
    //  B 4x16:  lane n = lane&15, k-pair offset = 2*(lane>>4)
    const int m_a  = lane & 15;
    const int koff = (lane >> 4) << 1;

    // prologue: stage block 0, wait own async loads, publish
    stage_frames_async(frames0, xs, 0, tid);
    asm volatile("s_wait_asynccnt 0x0" ::: "memory");
    __syncthreads();

    for (int blk = 0; blk < N_BLK; ++blk) {
        float* cur = (blk & 1) ? frames1 : frames0;
        float* nxt = (blk & 1) ? frames0 : frames1;

        // overlap: gather next block into the other buffer while computing
        if (blk + 1 < N_BLK) stage_frames_async(nxt, xs, blk + 1, tid);

        const int baseA0 = m_a * KSTRIDE + koff;
        const int baseA1 = baseA0 + 16 * KSTRIDE;

        // each wave owns N-tile pairs over 7 pairs (wave 7 idles)
        for (int np = wave; np < N_NPAIRS; np += 8) {
            const int baseB0 = (np * 32 + m_a) * KSTRIDE + koff;
            const int baseB1 = baseB0 + 16 * KSTRIDE;
            v8f c00 = {0.f,0.f,0.f,0.f,0.f,0.f,0.f,0.f};
            v8f c01 = c00, c10 = c00, c11 = c00;
#pragma unroll 4
            for (int kk = 0; kk < N_FFT / 4; ++kk) {
                v2f a0 = *(const v2f*)&cur[baseA0 + 4 * kk];
                v2f a1 = *(const v2f*)&cur[baseA1 + 4 * kk];
                v2f b0 = *(const v2f*)&basisT[baseB0 + 4 * kk];
                v2f b1 = *(const v2f*)&basisT[baseB1 + 4 * kk];
                c00 = __builtin_amdgcn_wmma_f32_16x16x4_f32(false, a0, false, b0, (short)0, c00, false, false);
                c01 = __builtin_amdgcn_wmma_f32_16x16x4_f32(false, a0, false, b1, (short)0, c01, false, false);
                c10 = __builtin_amdgcn_wmma_f32_16x16x4_f32(false, a1, false, b0, (short)0, c10, false, false);
                c11 = __builtin_amdgcn_wmma_f32_16x16x4_f32(false, a1, false, b1, (short)0, c11, false, false);
            }
            // C layout: vgpr r -> (frame r + 8*(lane>>4), col base + (lane&15)).
            // re (even col) / im (odd col) sit in adjacent lanes -> xor-merge.
            // c00/c10 (and c01/c11) share the same bins -> one atomic each.
            float ps0 = 0.0f, ps1 = 0.0f;
#pragma unroll
            for (int r = 0; r < 8; ++r) {
                float v, sq;
                v = c00[r]; sq = v * v; sq += __shfl_xor(sq, 1, 32); ps0 += sqrtf(sq);
                v = c10[r]; sq = v * v; sq += __shfl_xor(sq, 1, 32); ps0 += sqrtf(sq);
                v = c01[r]; sq = v * v; sq += __shfl_xor(sq, 1, 32); ps1 += sqrtf(sq);
                v = c11[r]; sq = v * v; sq += __shfl_xor(sq, 1, 32); ps1 += sqrtf(sq);
            }
            if ((lane & 1) == 0) {
                const int col0 = np * 32 + (lane & 15);
                const int col1 = col0 + 16;
                if (col0 < NCOLS) atomicAdd(&acc[col0 >> 1], ps0);  // ds_add_f32
                if (col1 < NCOLS) atomicAdd(&acc[col1 >> 1], ps1);
            }
        }

        // own async gathers done, then publish to all waves
        asm volatile("s_wait_asynccnt 0x0" ::: "memory");
        __syncthreads();
    }

    if (tid < NBINS) {
        mag_mean[(size_t)bc * NBINS + tid] = acc[tid] * INV_SCALE;
    }
}

// ============================================================
// Stage 2: band projections + embeddings (exact fp32, K <= 41)
//   out[((b*64+c)*5 + band)*128 + d]
// ============================================================
__global__ __launch_bounds__(256) void band_project_kernel(
    const float* __restrict__ mag_mean,
    const float* __restrict__ W0, const float* __restrict__ b0,
    const float* __restrict__ W1, const float* __restrict__ b1,
    const float* __restrict__ W2, const float* __restrict__ b2,
    const float* __restrict__ W3, const float* __restrict__ b3,
    const float* __restrict__ W4, const float* __restrict__ b4,
    const float* __restrict__ chemb, const float* __restrict__ bdemb,
    float* __restrict__ out)
{
    const int i = blockIdx.x * 256 + threadIdx.x;
    if (i >= BC * 5 * D_TOKEN) return;
    const int d    = i & (D_TOKEN - 1);
    const int rem  = i >> 7;
    const int band = rem % 5;
    const int bc   = rem / 5;
    const int ch   = bc & 63;

    const int   nb_tab[5]  = {7, 8, 10, 35, 41};
    const int   off_tab[5] = {0, 7, 15, 25, 60};
    const float* W_tab[5]  = {W0, W1, W2, W3, W4};
    const float* bv_tab[5] = {b0, b1, b2, b3, b4};

    const int    nb   = nb_tab[band];
    const float* feat = mag_mean + (size_t)bc * NBINS + off_tab[band];
    const float* W    = W_tab[band];

    float accv = bv_tab[band][d] + chemb[ch * D_TOKEN + d] + bdemb[band * D_TOKEN + d];
    for (int j = 0; j < nb; ++j) {
        accv = fmaf(feat[j], W[j * D_TOKEN + d], accv);
    }
    out[i] = accv;
}

// ============================================================
extern "C" void kernel_launch(void* const* d_in, const int* in_sizes, int n_in,
                              void* d_out, int out_size, void* d_ws, size_t ws_size,
                              hipStream_t stream) {
    const float* x     = (const float*)d_in[0];
    const float* Wd    = (const float*)d_in[1];
    const float* bd    = (const float*)d_in[2];
    const float* Wt    = (const float*)d_in[3];
    const float* bt    = (const float*)d_in[4];
    const float* Wa    = (const float*)d_in[5];
    const float* ba    = (const float*)d_in[6];
    const float* Wb    = (const float*)d_in[7];
    const float* bb    = (const float*)d_in[8];
    const float* Wg    = (const float*)d_in[9];
    const float* bg    = (const float*)d_in[10];
    const float* chemb = (const float*)d_in[11];
    const float* bdemb = (const float*)d_in[12];
    float* out = (float*)d_out;
    float* mag_mean = (float*)d_ws;   // 1024*101 floats

    const size_t lds_bytes =
        (size_t)(NCOLS_PAD * KSTRIDE + 2 * 32 * KSTRIDE + 256 + 104) * sizeof(float);

    stft_magmean_wmma_kernel<<<BC, 256, lds_bytes, stream>>>(x, mag_mean);

    const int n_out = BC * 5 * D_TOKEN;
    band_project_kernel<<<(n_out + 255) / 256, 256, 0, stream>>>(
        mag_mean, Wd, bd, Wt, bt, Wa, ba, Wb, bb, Wg, bg, chemb, bdemb, out);
}